// VectorQuantizer_42082089566544
// MI455X (gfx1250) — compile-verified
//
#include <hip/hip_runtime.h>
#include <hip/hip_bf16.h>

// Problem constants (match reference)
#define BB 32
#define DD 64
#define TT 4096
#define KK 512
#define NN (BB * TT)          // 131072 points
#define ROWS 66               // 64 dwords + 2-dword TDM pad -> conflict-free b64 reads

typedef float v2f __attribute__((ext_vector_type(2)));
typedef float v8f __attribute__((ext_vector_type(8)));
typedef unsigned int u32x4 __attribute__((ext_vector_type(4)));
typedef unsigned int u32x8 __attribute__((ext_vector_type(8)));

// ---------------------------------------------------------------------------
// Kernel 1: nearest-code search via V_WMMA_F32_16X16X4_F32.
//   256 threads (8 waves). Wave 0 stages the codebook into LDS with one
//   TENSOR_LOAD_TO_LDS (TDM inserts 2 pad dwords per 64-dword row -> LDS
//   layout scb[n*66 + d]); each wave then owns one 16-point M tile.
// ---------------------------------------------------------------------------
__global__ void vq_argmin_wmma_kernel(const float* __restrict__ z,
                                      const float* __restrict__ cb,
                                      int* __restrict__ idx_ws,
                                      float* __restrict__ idx_out) {
    extern __shared__ float smem[];
    float* scb    = smem;               // [KK][ROWS] padded row-major codebook
    float* senorm = smem + KK * ROWS;   // [KK] ||e||^2

    const int tid = threadIdx.x;

    // ---- TDM: DMA codebook (512x64 f32) into LDS with row padding ----------
    if (tid < 32) {                                  // wave 0 only (one issue)
        unsigned long long ga = (unsigned long long)(uintptr_t)cb;
        u32x4 g0;
        g0.x = 0x00000001u;                          // count=1, load, no gather
        g0.y = 0u;                                   // lds_addr = 0 (dyn-LDS base)
        g0.z = (unsigned int)(ga & 0xffffffffu);     // global_addr[31:0]
        g0.w = (unsigned int)((ga >> 32) & 0x01ffffffu) | (2u << 30); // [56:32]|type=2

        u32x8 g1;
        g1.s0 = (2u << 16)                           // data_size = 4 bytes
              | (1u << 20)                           // pad_enable
              | (5u << 22)                           // pad_interval: 64 dwords
              | (1u << 25);                          // pad_amount: 2 dwords
        g1.s1 = (unsigned int)DD << 16;              // tensor_dim0 = 64 (lo16)
        g1.s2 = (unsigned int)KK << 16;              // tensor_dim1 = 512 (lo16)
        g1.s3 = (unsigned int)DD << 16;              // tile_dim0 = 64
        g1.s4 = (unsigned int)KK;                    // tile_dim1 = 512
        g1.s5 = (unsigned int)DD;                    // tensor_dim0_stride = 64
        g1.s6 = 0u;
        g1.s7 = 0u;

        asm volatile("tensor_load_to_lds %0, %1"
                     :
                     : "s"(g0), "s"(g1)
                     : "memory");
        __builtin_amdgcn_s_wait_tensorcnt(0);
    }
    __syncthreads();                                 // publish LDS to all waves

    // ---- ||e||^2 per code -------------------------------------------------
    for (int n = tid; n < KK; n += 256) {
        const float* row = scb + n * ROWS;
        float s = 0.f;
        #pragma unroll 8
        for (int d = 0; d < DD; ++d) { float v = row[d]; s += v * v; }
        senorm[n] = s;
    }
    __syncthreads();

    const int wave  = tid >> 5;
    const int lane  = tid & 31;
    const int m     = lane & 15;   // A row (point) / C column (code)
    const int h     = lane >> 4;   // half: A K-offset 2h, C rows h*8+r
    const int tileM = blockIdx.x * 8 + wave;
    const int t0    = (tileM * 16) % TT;
    const int b     = (tileM * 16) / TT;             // TT%16==0: no straddle

    // A tile (16 pts x 64 dims), ISA 16x4 f32 layout per k-step:
    // lanes 0-15 hold K=0,1 ; lanes 16-31 hold K=2,3.
    const float* zb = z + (size_t)b * DD * TT + t0 + m;
    v2f a[16];
    #pragma unroll
    for (int k = 0; k < 16; ++k) {
        a[k].x = zb[(size_t)(4 * k + 2 * h + 0) * TT];
        a[k].y = zb[(size_t)(4 * k + 2 * h + 1) * TT];
    }

    float bestS[8];
    int   bestI[8];
    #pragma unroll
    for (int r = 0; r < 8; ++r) { bestS[r] = 3.4e38f; bestI[r] = 0; }

    for (int ct = 0; ct < KK / 16; ++ct) {           // 32 code tiles
        const int   n  = ct * 16 + m;                // this lane's code column
        const float en = senorm[n];

        // B pairs are adjacent, 8B-aligned: single ds_load_b64 per k-step,
        // all 16 from one base with immediate offsets (<= 248 B).
        const v2f* brow = (const v2f*)(scb + n * ROWS);

        v8f c0 = {}, c1 = {};                        // two independent chains
        #pragma unroll
        for (int k = 0; k < 16; k += 2) {
            v2f bf0 = brow[2 * k + h];
            v2f bf1 = brow[2 * (k + 1) + h];
            c0 = __builtin_amdgcn_wmma_f32_16x16x4_f32(
                     false, a[k],     false, bf0, (short)0, c0, false, false);
            c1 = __builtin_amdgcn_wmma_f32_16x16x4_f32(
                     false, a[k + 1], false, bf1, (short)0, c1, false, false);
        }

        #pragma unroll
        for (int r = 0; r < 8; ++r) {
            float sc = en - 2.0f * (c0[r] + c1[r]);  // ||z||^2 const per row
            if (sc < bestS[r]) { bestS[r] = sc; bestI[r] = n; }
        }
    }

    // Butterfly min across the 16 lanes sharing the same row set.
    #pragma unroll
    for (int mask = 1; mask <= 8; mask <<= 1) {
        #pragma unroll
        for (int r = 0; r < 8; ++r) {
            float s2 = __shfl_xor(bestS[r], mask, 32);
            int   i2 = __shfl_xor(bestI[r], mask, 32);
            if (s2 < bestS[r] || (s2 == bestS[r] && i2 < bestI[r])) {
                bestS[r] = s2; bestI[r] = i2;
            }
        }
    }

    if (m == 0) {                                    // lanes 0 & 16: 8 rows each
        #pragma unroll
        for (int r = 0; r < 8; ++r) {
            int p = tileM * 16 + h * 8 + r;
            idx_ws[p]  = bestI[r];
            idx_out[p] = (float)bestI[r];
        }
    }
}

// ---------------------------------------------------------------------------
// Kernel 2: gather codebook rows -> quantized [B,D,T]; block partial sums of
// (q - z)^2 into workspace.
// ---------------------------------------------------------------------------
__global__ void vq_gather_kernel(const float* __restrict__ z,
                                 const float* __restrict__ cb,
                                 const int* __restrict__ idx_ws,
                                 float* __restrict__ out_q,
                                 float* __restrict__ partial) {
    const int tid = threadIdx.x;
    const int p   = blockIdx.x * 256 + tid;
    const int b   = p >> 12;                         // / TT
    const int t   = p & (TT - 1);
    const int idx = idx_ws[p];

    const float4* cb4 = (const float4*)(cb + (size_t)idx * DD);
    const float*  zb  = z     + (size_t)b * DD * TT + t;
    float*        qb  = out_q + (size_t)b * DD * TT + t;

    float s = 0.f;
    #pragma unroll
    for (int d4 = 0; d4 < DD / 4; ++d4) {
        float4 q4 = cb4[d4];
        float qv[4] = {q4.x, q4.y, q4.z, q4.w};
        #pragma unroll
        for (int j = 0; j < 4; ++j) {
            int d = d4 * 4 + j;
            float zv = zb[(size_t)d * TT];
            qb[(size_t)d * TT] = qv[j];              // coalesced along t
            float df = qv[j] - zv;
            s += df * df;
        }
    }

    __shared__ float red[256];
    red[tid] = s;
    __syncthreads();
    for (int st = 128; st > 0; st >>= 1) {
        if (tid < st) red[tid] += red[tid + st];
        __syncthreads();
    }
    if (tid == 0) partial[blockIdx.x] = red[0];
}

// ---------------------------------------------------------------------------
// Kernel 3: deterministic final reduction -> vq_loss = 1.25 * mean((q-z)^2)
// ---------------------------------------------------------------------------
__global__ void vq_loss_kernel(const float* __restrict__ partial,
                               float* __restrict__ out_loss) {
    const int tid = threadIdx.x;
    float s = partial[tid] + partial[tid + 256];     // 512 partials
    __shared__ float red[256];
    red[tid] = s;
    __syncthreads();
    for (int st = 128; st > 0; st >>= 1) {
        if (tid < st) red[tid] += red[tid + st];
        __syncthreads();
    }
    if (tid == 0) {
        out_loss[0] = 1.25f * red[0] / (float)((size_t)BB * DD * TT);
    }
}

extern "C" void kernel_launch(void* const* d_in, const int* in_sizes, int n_in,
                              void* d_out, int out_size, void* d_ws, size_t ws_size,
                              hipStream_t stream) {
    (void)in_sizes; (void)n_in; (void)out_size; (void)ws_size;

    const float* z  = (const float*)d_in[0];   // [B, D, T]
    const float* cb = (const float*)d_in[1];   // [K, D]

    float* out      = (float*)d_out;
    float* out_loss = out;                               // [1]
    float* out_q    = out + 1;                           // [B, D, T]
    float* out_idx  = out + 1 + (size_t)BB * DD * TT;    // [B, T] as float

    int*   idx_ws  = (int*)d_ws;                         // N ints
    float* partial = (float*)d_ws + NN;                  // 512 floats

    const size_t shmem = (size_t)(KK * ROWS + KK) * sizeof(float);

    vq_argmin_wmma_kernel<<<NN / 128, 256, shmem, stream>>>(z, cb, idx_ws, out_idx);
    vq_gather_kernel<<<NN / 256, 256, 0, stream>>>(z, cb, idx_ws, out_q, partial);
    vq_loss_kernel<<<1, 256, 0, stream>>>(partial, out_loss);
}